// Model_24232205484232
// MI455X (gfx1250) — compile-verified
//
#include <hip/hip_runtime.h>
#include <stdint.h>

// ---------------- problem constants (from reference) ----------------
#define NB      8
#define NA      65280           // anchors
#define NA4     (NA / 4)        // 16320 float4 groups
#define MAXDET  300
#define BLK     1024
#define NCHUNK  ((NA + BLK - 1) / BLK)   // 64 suppression chunks

constexpr float IOU_T   = 0.4f;
constexpr float SCORE_T = 1e-8f;
constexpr float NEGV    = -1e9f;
constexpr float IMG     = 1024.0f;

// ---------------- CDNA5 async global->LDS DMA helpers ----------------
// GVS mode: mem_addr = SGPR_base + VGPR_offset(i32) ; LDS dest addr comes
// from a VGPR (wave-relative byte offset). Tracked on ASYNCcnt.
__device__ __forceinline__ void async_ld_b128(uint32_t lds_byte_addr,
                                              int      glb_byte_off,
                                              const void* sbase) {
  asm volatile("global_load_async_to_lds_b128 %0, %1, %2"
               :: "v"(lds_byte_addr), "v"(glb_byte_off), "s"(sbase)
               : "memory");
}
#define S_WAIT_ASYNC(n) asm volatile("s_wait_asynccnt " #n ::: "memory")

__device__ __forceinline__ uint32_t lds_addr_of(const void* p) {
  // generic pointers into LDS carry the wave-relative LDS offset in the
  // low 32 bits (flat-LDS aperture mapping), which is what the async
  // instruction's VDST expects.
  return (uint32_t)(uintptr_t)p;
}

// ---------------- kernel 1: box decode + softmax score ----------------
__global__ __launch_bounds__(256) void decode_kernel(
    const float*  __restrict__ enc,      // [B,N,4]
    const float*  __restrict__ logits,   // [B,N,2]
    const float*  __restrict__ anchors,  // [N,4]
    float4*       __restrict__ ws_boxes, // [B*N]
    float*        __restrict__ ws_scores)// [B*N]
{
  int g = blockIdx.x * 256 + threadIdx.x;
  if (g >= NB * NA) return;
  int n = g % NA;

  const float4 e = ((const float4*)enc)[g];
  const float4 a = ((const float4*)anchors)[n];
  const float2 l = ((const float2*)logits)[g];

  float ycen_a = (a.x + a.z) * 0.5f;
  float xcen_a = (a.y + a.w) * 0.5f;
  float ha = a.z - a.x;
  float wa = a.w - a.y;

  float ycen = (e.x * 0.1f) * ha + ycen_a;      // scale 10
  float xcen = (e.y * 0.1f) * wa + xcen_a;      // scale 10
  float h    = expf(e.z * 0.2f) * ha;           // scale 5
  float w    = expf(e.w * 0.2f) * wa;           // scale 5

  float4 b;
  b.x = fminf(fmaxf(ycen - 0.5f * h, 0.0f), IMG);
  b.y = fminf(fmaxf(xcen - 0.5f * w, 0.0f), IMG);
  b.z = fminf(fmaxf(ycen + 0.5f * h, 0.0f), IMG);
  b.w = fminf(fmaxf(xcen + 0.5f * w, 0.0f), IMG);

  ws_boxes[g]  = b;
  ws_scores[g] = 1.0f / (1.0f + expf(l.x - l.y));  // softmax[...,1]
}

// ---------------- kernel 2: persistent greedy NMS, 1 block / batch ----
__global__ __launch_bounds__(BLK) void nms_kernel(
    const float*  __restrict__ angle_pred, // [B,N,3]
    const float4* __restrict__ ws_boxes,   // [B*N]
    const float*  __restrict__ ws_scores,  // [B*N]
    float*        __restrict__ out)        // boxes|scores|angles|num
{
  __shared__ float  s_live[NA];          // 255 KB live scores (CDNA5 LDS)
  __shared__ float4 s_tile[2][BLK];      // 32 KB double-buffered box tiles
  __shared__ float  s_rv[32];
  __shared__ int    s_ri[32];
  __shared__ float4 s_bbox;
  __shared__ int    s_valid;

  const int b    = blockIdx.x;
  const int tid  = threadIdx.x;
  const int lane = tid & 31;
  const int wave = tid >> 5;
  const size_t base = (size_t)b * NA;
  const float4* bboxes  = ws_boxes  + base;
  const float*  bscores = ws_scores + base;

  float* det_boxes  = out;                        // [B,300,4]
  float* det_scores = out + NB * MAXDET * 4;      // [B,300]
  float* det_angles = out + NB * MAXDET * 5;      // [B,300,3]
  float* num_det    = out + NB * MAXDET * 8;      // [B]

  // ---- stage live scores into LDS with async DMA (B128 per lane) ----
  {
    float4* live4 = (float4*)s_live;
    #pragma unroll
    for (int c = 0; c < 16; ++c) {
      int i4 = c * BLK + tid;
      if (i4 < NA4) {
        async_ld_b128(lds_addr_of(&live4[i4]), i4 * 16, (const void*)bscores);
      }
    }
    S_WAIT_ASYNC(0);
  }
  __syncthreads();

  const uint32_t lds_t0 = lds_addr_of(&s_tile[0][tid]);
  const uint32_t lds_t1 = lds_addr_of(&s_tile[1][tid]);

  int count = 0;

  for (int it = 0; it < MAXDET; ++it) {
    // ---------------- parallel argmax over live scores ----------------
    float bv = -3.4e38f;
    int   bi = 0;
    {
      const float4* live4 = (const float4*)s_live;
      #pragma unroll
      for (int c = 0; c < 16; ++c) {
        int i4 = c * BLK + tid;
        if (i4 < NA4) {
          float4 v = live4[i4];
          int ib = i4 * 4;
          if (v.x > bv) { bv = v.x; bi = ib;     }
          if (v.y > bv) { bv = v.y; bi = ib + 1; }
          if (v.z > bv) { bv = v.z; bi = ib + 2; }
          if (v.w > bv) { bv = v.w; bi = ib + 3; }
        }
      }
    }
    // wave32 reduction, lowest index wins ties (matches jnp.argmax)
    #pragma unroll
    for (int off = 16; off > 0; off >>= 1) {
      float ov = __shfl_xor(bv, off, 32);
      int   oi = __shfl_xor(bi, off, 32);
      if (ov > bv || (ov == bv && oi < bi)) { bv = ov; bi = oi; }
    }
    if (lane == 0) { s_rv[wave] = bv; s_ri[wave] = bi; }
    __syncthreads();

    if (wave == 0) {
      bv = s_rv[lane];
      bi = s_ri[lane];
      #pragma unroll
      for (int off = 16; off > 0; off >>= 1) {
        float ov = __shfl_xor(bv, off, 32);
        int   oi = __shfl_xor(bi, off, 32);
        if (ov > bv || (ov == bv && oi < bi)) { bv = ov; bi = oi; }
      }
      if (lane == 0) {
        int valid = (bv > SCORE_T) ? 1 : 0;
        s_valid = valid;
        float4 bb = bboxes[bi];   // L2-resident
        s_bbox = bb;
        float* ob = det_boxes  + ((size_t)b * MAXDET + it) * 4;
        float* oa = det_angles + ((size_t)b * MAXDET + it) * 3;
        if (valid) {
          ob[0] = bb.x; ob[1] = bb.y; ob[2] = bb.z; ob[3] = bb.w;
          det_scores[b * MAXDET + it] = bv;
          const float* ap = angle_pred + (base + (size_t)bi) * 3;
          oa[0] = ap[0]; oa[1] = ap[1]; oa[2] = ap[2];
          ++count;
        } else {
          ob[0] = 0.f; ob[1] = 0.f; ob[2] = 0.f; ob[3] = 0.f;
          det_scores[b * MAXDET + it] = 0.f;
          oa[0] = 0.f; oa[1] = 0.f; oa[2] = 0.f;
        }
      }
    }
    __syncthreads();

    // ---------------- IoU suppression sweep (async pipelined) ----------
    if (s_valid) {
      const float4 bb = s_bbox;
      const float barea = (bb.z - bb.x) * (bb.w - bb.y);

      // prologue: chunk 0 -> buf0
      {
        int i0 = tid; if (i0 > NA - 1) i0 = NA - 1;
        async_ld_b128(lds_t0, i0 * 16, (const void*)bboxes);
      }
      for (int c = 0; c < NCHUNK; ++c) {
        // prefetch chunk c+1 into the other buffer (clamped re-issue on tail)
        int inext = (c + 1) * BLK + tid;
        if (inext > NA - 1) inext = NA - 1;
        async_ld_b128((c & 1) ? lds_t0 : lds_t1, inext * 16,
                      (const void*)bboxes);
        S_WAIT_ASYNC(1);                 // chunk c is now in LDS

        float4 bx = s_tile[c & 1][tid];
        int i = c * BLK + tid;
        if (i < NA) {
          float iy = fmaxf(fminf(bb.z, bx.z) - fmaxf(bb.x, bx.x), 0.0f);
          float ix = fmaxf(fminf(bb.w, bx.w) - fmaxf(bb.y, bx.y), 0.0f);
          float inter = iy * ix;
          float area  = (bx.z - bx.x) * (bx.w - bx.y);
          float iou   = inter / fmaxf(area + barea - inter, 1e-8f);
          if (iou > IOU_T) s_live[i] = NEGV;   // suppress (incl. best itself)
        }
      }
      S_WAIT_ASYNC(0);                   // drain before buffers are reused
    }
    __syncthreads();
  }

  if (tid == 0) num_det[b] = (float)count;
}

// ---------------- host launcher ----------------
extern "C" void kernel_launch(void* const* d_in, const int* in_sizes, int n_in,
                              void* d_out, int out_size, void* d_ws, size_t ws_size,
                              hipStream_t stream) {
  const float* enc     = (const float*)d_in[0];  // box_encodings  [8,65280,4]
  const float* logits  = (const float*)d_in[1];  // objectness     [8,65280,2]
  const float* angles  = (const float*)d_in[2];  // angle_pred     [8,65280,3]
  const float* anchors = (const float*)d_in[3];  // anchors        [65280,4]

  float4* ws_boxes  = (float4*)d_ws;                          // B*N float4
  float*  ws_scores = (float*)d_ws + (size_t)NB * NA * 4;     // B*N floats

  float* out = (float*)d_out;

  int total = NB * NA;
  decode_kernel<<<(total + 255) / 256, 256, 0, stream>>>(
      enc, logits, anchors, ws_boxes, ws_scores);

  nms_kernel<<<NB, BLK, 0, stream>>>(angles, ws_boxes, ws_scores, out);
}